// YoloLoss_37117107372543
// MI455X (gfx1250) — compile-verified
//
#include <hip/hip_runtime.h>
#include <hip/hip_bf16.h>
#include <math.h>

// ---------------------------------------------------------------------------
// YOLO loss for MI455X (gfx1250, wave32).
//   Kernel A (1 wave, serial):  replay the taken-map assignment scan exactly
//                               (taken carries across images, per reference),
//                               accumulate positive-sample losses, spill per-cell
//                               "first-taken-at-image" stamps LDS->global via
//                               global_store_async_from_lds_b32 (ASYNCcnt).
//   Kernel B (1200 blocks):     per-cell no-obj loss; GT boxes staged into LDS
//                               with one wave32 global_load_async_to_lds_b128;
//                               WMMA f32 16x16x4 as the wave32 sum reducer.
//   Kernel C (1 wave):          deterministic final sum (WMMA reduce), /B.
// ---------------------------------------------------------------------------

#define NCLASSES 80
#define LCLASS   1.0f
#define LOBJ     1.0f
#define LBOX     10.0f
#define IOU_TH   0.5f
#define NB_      16          // batch
#define NA_      3           // anchors
#define NH_      80
#define NW_      80
#define NBOX_    32
#define DD_      (5 + NCLASSES)       // 85
#define CELLS_PER_IMG (NA_ * NH_ * NW_)   // 19200
#define BLOCKS_PER_IMG (CELLS_PER_IMG / 256) // 75
#define NBLK_    (NB_ * BLOCKS_PER_IMG)      // 1200

typedef __attribute__((ext_vector_type(2))) float v2f;
typedef __attribute__((ext_vector_type(8))) float v8f;

__device__ __forceinline__ float softplus_f(float x) {
    // log(1 + e^x), stable: max(x,0) + log1p(exp(-|x|))
    return fmaxf(x, 0.0f) + log1pf(expf(-fabsf(x)));
}

// Wave32 sum via V_WMMA_F32_16X16X4_F32 (confirmed lowering in round 1).
// A (16x4 f32): lanes 0-15 hold K=0,1; lanes 16-31 hold K=2,3.
// a={val,0}, B=ones(4x16)  =>  D[m][n] = val[m] + val[m+16] for all n.
// Lane L holds column N=L%16, rows 0..7 (L<16) or 8..15 (L>=16) in 8 VGPRs;
// summing them + one shfl_xor(16) yields the exact f32 32-lane sum.
__device__ __forceinline__ float wave_sum_wmma(float val) {
    v2f a; a.x = val;  a.y = 0.0f;
    v2f b; b.x = 1.0f; b.y = 1.0f;
    v8f c = {};
    c = __builtin_amdgcn_wmma_f32_16x16x4_f32(
            /*neg_a=*/false, a, /*neg_b=*/false, b,
            /*c_mod=*/(short)0, c, /*reuse_a=*/false, /*reuse_b=*/false);
    float s = c[0] + c[1] + c[2] + c[3] + c[4] + c[5] + c[6] + c[7];
    s += __shfl_xor(s, 16, 32);
    return s;
}

// ---------------------------------------------------------------------------
// Kernel A: serial assignment replay (single wave32).
// ---------------------------------------------------------------------------
__global__ __launch_bounds__(32)
void yolo_assign(const float* __restrict__ X,
                 const float* __restrict__ yboxes,
                 const int*   __restrict__ ylabels,
                 const float* __restrict__ anchors,
                 unsigned char* __restrict__ stamps_g,
                 float* __restrict__ posLoss) {
    __shared__ __align__(16) unsigned char stamp[CELLS_PER_IMG];
    const int lane = threadIdx.x;

    for (int i = lane; i < CELLS_PER_IMG; i += 32) stamp[i] = 0xFF;
    __syncthreads();

    float aw[NA_], ah[NA_];
    #pragma unroll
    for (int k = 0; k < NA_; ++k) { aw[k] = anchors[2*k]; ah[k] = anchors[2*k+1]; }

    float pos = 0.0f;   // accumulated on lane 0 only

    for (int b = 0; b < NB_; ++b) {
        for (int i = 0; i < NBOX_; ++i) {
            const float bx = yboxes[(b*NBOX_ + i)*4 + 0];
            const float by = yboxes[(b*NBOX_ + i)*4 + 1];
            const float bw = yboxes[(b*NBOX_ + i)*4 + 2];
            const float bh = yboxes[(b*NBOX_ + i)*4 + 3];
            const int  lab = ylabels[b*NBOX_ + i];

            int y = (int)floorf(by * (float)NH_); y = min(max(y, 0), NH_ - 1);
            int x = (int)floorf(bx * (float)NW_); x = min(max(x, 0), NW_ - 1);
            const float yrel = (by - (float)y / (float)NH_) * (float)NH_;
            const float xrel = (bx - (float)x / (float)NW_) * (float)NW_;

            // wh-IoU vs anchors
            float iou[NA_];
            #pragma unroll
            for (int k = 0; k < NA_; ++k) {
                float inter = fminf(bw, aw[k]) * fminf(bh, ah[k]);
                float uni   = bw*bh + aw[k]*ah[k] - inter;
                iou[k] = inter / (uni + 1e-9f);
            }
            // stable descending argsort of 3 (matches jnp.argsort(-iou))
            int ord[3] = {0, 1, 2};
            if (iou[ord[0]] < iou[ord[1]]) { int t=ord[0]; ord[0]=ord[1]; ord[1]=t; }
            if (iou[ord[1]] < iou[ord[2]]) { int t=ord[1]; ord[1]=ord[2]; ord[2]=t; }
            if (iou[ord[0]] < iou[ord[1]]) { int t=ord[0]; ord[0]=ord[1]; ord[1]=t; }

            int anch = -1;
            #pragma unroll
            for (int k = 0; k < NA_; ++k) {
                int a2 = ord[k];
                if (anch < 0 && stamp[a2*(NH_*NW_) + y*NW_ + x] == 0xFF) anch = a2;
            }
            const bool has = (anch >= 0);   // uniform across the wave

            if (has) {
                if (lane == 0) stamp[anch*(NH_*NW_) + y*NW_ + x] = (unsigned char)b;

                const float* p = X + (size_t)((((b*NA_ + anch)*NH_ + y)*NW_ + x)) * DD_;
                __builtin_prefetch(p, 0, 0);   // global_prefetch_b8

                // lane-parallel logsumexp over 80 class logits
                const int i0 = lane, i1 = lane + 32, i2 = lane + 64;
                float v0 = (i0 < NCLASSES) ? p[5 + i0] : -INFINITY;
                float v1 = (i1 < NCLASSES) ? p[5 + i1] : -INFINITY;
                float v2 = (i2 < NCLASSES) ? p[5 + i2] : -INFINITY;
                float m = fmaxf(v0, fmaxf(v1, v2));
                #pragma unroll
                for (int off = 16; off > 0; off >>= 1)
                    m = fmaxf(m, __shfl_xor(m, off, 32));
                float e = expf(v0 - m) + expf(v1 - m) + expf(v2 - m);
                #pragma unroll
                for (int off = 16; off > 0; off >>= 1)
                    e += __shfl_xor(e, off, 32);
                const float lse = logf(e) + m;

                if (lane == 0) {
                    const float p0 = p[0];
                    const float obj_l = LOBJ * (softplus_f(p0) - p0);   // bce(p0, 1)
                    const float t2 = logf(bw / aw[anch] + 1e-16f);
                    const float t3 = logf(bh / ah[anch] + 1e-16f);
                    const float d0 = p[1] - xrel, d1 = p[2] - yrel;
                    const float d2 = p[3] - t2,   d3 = p[4] - t3;
                    const float box_l = LBOX * 0.25f * (d0*d0 + d1*d1 + d2*d2 + d3*d3);
                    const float cls_l = LCLASS * (lse - p[5 + lab]);
                    pos += obj_l + box_l + cls_l;
                }
            }
        }
    }

    __syncthreads();   // drains DScnt: LDS stamp writes are committed

    // Spill the stamp map LDS -> global with async stores (ASYNCcnt path).
    // 32 lanes x b32 = 128 B per iteration; 19200/128 = 150 iterations.
    {
        const unsigned ldsBase = (unsigned)(uintptr_t)(&stamp[0]);
        for (int i = lane * 4; i < CELLS_PER_IMG; i += 128) {
            unsigned ldsa = ldsBase + (unsigned)i;
            unsigned long long ga = (unsigned long long)(uintptr_t)(stamps_g + i);
            asm volatile("global_store_async_from_lds_b32 %0, %1, off"
                         :: "v"(ga), "v"(ldsa) : "memory");
        }
        asm volatile("s_wait_asynccnt 0" ::: "memory");
    }

    if (lane == 0) posLoss[0] = pos;
}

// ---------------------------------------------------------------------------
// Kernel B: no-obj loss; one block of 256 = 8 waves per 256 cells of one image.
// ---------------------------------------------------------------------------
__global__ __launch_bounds__(256)
void yolo_noobj(const float* __restrict__ X,
                const float* __restrict__ yboxes,
                const float* __restrict__ anchors,
                const unsigned char* __restrict__ stamps,
                float* __restrict__ partials) {
    __shared__ __align__(16) float boxs[NBOX_ * 4];   // 512 B
    __shared__ float wred[8];

    const int b = blockIdx.x / BLOCKS_PER_IMG;
    const int r = blockIdx.x % BLOCKS_PER_IMG;

    // Stage this image's 32 boxes (512 B) into LDS with one wave32 async b128:
    // 32 lanes x 16 B, bypassing VGPRs entirely.
    if (threadIdx.x < 32) {
        unsigned ldsa = (unsigned)(uintptr_t)(&boxs[0]) + threadIdx.x * 16u;
        unsigned long long ga = (unsigned long long)(uintptr_t)
                                (yboxes + (size_t)b * (NBOX_ * 4) + threadIdx.x * 4);
        asm volatile("global_load_async_to_lds_b128 %0, %1, off"
                     :: "v"(ldsa), "v"(ga) : "memory");
        asm volatile("s_wait_asynccnt 0" ::: "memory");
    }
    __syncthreads();

    const int cell = r * 256 + threadIdx.x;        // a*6400 + y*80 + x
    const int a    = cell / (NH_ * NW_);

    const float* p = X + (size_t)(b * CELLS_PER_IMG + cell) * DD_;
    __builtin_prefetch(p, 0, 0);

    const float p0 = p[0], px = p[1], py = p[2], pw = p[3], ph = p[4];
    const float aw = anchors[2*a], ah = anchors[2*a + 1];
    const float sw = pw * aw, sh = ph * ah;

    const float x11 = px - sw * 0.5f, x12 = px + sw * 0.5f;
    const float y11 = py - sh * 0.5f, y12 = py + sh * 0.5f;
    const float area1 = (x12 - x11) * (y12 - y11);

    float maxiou = -INFINITY;
    #pragma unroll 4
    for (int j = 0; j < NBOX_; ++j) {
        const float gx = boxs[4*j+0], gy = boxs[4*j+1];
        const float gw = boxs[4*j+2], gh = boxs[4*j+3];
        const float x21 = gx - gw * 0.5f, x22 = gx + gw * 0.5f;
        const float y21 = gy - gh * 0.5f, y22 = gy + gh * 0.5f;
        const float iw = fmaxf(fminf(x12, x22) - fmaxf(x11, x21), 0.0f);
        const float ih = fmaxf(fminf(y12, y22) - fmaxf(y11, y21), 0.0f);
        const float inter = iw * ih;
        const float uni = area1 + (x22 - x21) * (y22 - y21) - inter;
        maxiou = fmaxf(maxiou, inter / (uni + 1e-9f));
    }

    // taken==0 at image b  <=>  first-taken image stamp > b (255 = never)
    const bool notTaken = ((int)stamps[cell]) > b;
    const float val = (notTaken && (maxiou < IOU_TH)) ? softplus_f(p0) : 0.0f;

    // wave32 sum via WMMA (EXEC all ones here: no divergence at this point)
    const float wsum = wave_sum_wmma(val);
    if ((threadIdx.x & 31) == 0) wred[threadIdx.x >> 5] = wsum;
    __syncthreads();
    if (threadIdx.x == 0) {
        float s = 0.0f;
        #pragma unroll
        for (int k = 0; k < 8; ++k) s += wred[k];
        partials[blockIdx.x] = s;
    }
}

// ---------------------------------------------------------------------------
// Kernel C: deterministic final reduction (fixed lane->index mapping + fixed
// WMMA reduction tree => bitwise-stable across replays).
// ---------------------------------------------------------------------------
__global__ __launch_bounds__(32)
void yolo_final(const float* __restrict__ partials,
                const float* __restrict__ posLoss,
                float* __restrict__ out) {
    const int lane = threadIdx.x;
    float s = 0.0f;
    for (int i = lane; i < NBLK_; i += 32) s += partials[i];
    const float tot = wave_sum_wmma(s);
    if (lane == 0)
        out[0] = (float)(((double)posLoss[0] + (double)LOBJ * (double)tot) / (double)NB_);
}

// ---------------------------------------------------------------------------
extern "C" void kernel_launch(void* const* d_in, const int* in_sizes, int n_in,
                              void* d_out, int out_size, void* d_ws, size_t ws_size,
                              hipStream_t stream) {
    const float* X       = (const float*)d_in[0];
    const float* yboxes  = (const float*)d_in[1];
    const int*   ylabels = (const int*)  d_in[2];
    const float* anchors = (const float*)d_in[3];
    float* out = (float*)d_out;

    unsigned char* stamps = (unsigned char*)d_ws;                       // 19200 B
    float* partials = (float*)((char*)d_ws + 19456);                    // 1200 f
    float* posLoss  = (float*)((char*)d_ws + 19456 + NBLK_ * 4);        // 1 f

    yolo_assign<<<1, 32, 0, stream>>>(X, yboxes, ylabels, anchors, stamps, posLoss);
    yolo_noobj <<<NBLK_, 256, 0, stream>>>(X, yboxes, anchors, stamps, partials);
    yolo_final <<<1, 32, 0, stream>>>(partials, posLoss, out);
}